// dmc_23046794510620
// MI455X (gfx1250) — compile-verified
//
#include <hip/hip_runtime.h>

// ============================================================================
// Muskingum-Cunge tree routing, persistent-kernel design for MI455X (gfx1250).
//
// Latency-bound problem: 336 sequential steps x 17 dependent tree levels.
// Strategy: keep ALL per-node state resident in LDS (320KB/WGP on CDNA5),
// one grid-wide barrier per step, async global->LDS prefetch of the next
// q_prime row (ASYNCcnt path) overlapped with the elementwise math.
// ============================================================================

#define NBLK   64
#define NTHR   256
#define NSTEPS 335          // scan consumes q_prime rows 0..334
#define NNODES 131071
#define Q_LB_F 1.0e-4f

// --- gfx1250 async global->LDS path (guarded; falls back to sync copy) ------
#if defined(__has_builtin)
#  if __has_builtin(__builtin_amdgcn_global_load_async_to_lds_b32)
#    define HAS_ASYNC 1
#  endif
#endif
#ifndef HAS_ASYNC
#  define HAS_ASYNC 0
#endif

typedef __attribute__((address_space(1))) int g_i32;
typedef __attribute__((address_space(3))) int l_i32;

__device__ __forceinline__ void async_f32(const float* g, float* l) {
#if HAS_ASYNC
  // prototype (from clang diagnostic): (addrspace(1) int*, addrspace(3) int*, imm, imm)
  __builtin_amdgcn_global_load_async_to_lds_b32((g_i32*)g, (l_i32*)l, 0, 0);
#else
  *l = *g;
#endif
}

__device__ __forceinline__ void wait_async_lds() {
#if HAS_ASYNC
  asm volatile("s_wait_asynccnt 0" ::: "memory");
#endif
}

// --- local (per-block) tree slice mapping -----------------------------------
// Block owns, for level l in 0..6, SIZES[l]/64 = 1<<(10-l) contiguous nodes.
// Local index li in [0,2032): levels packed at LOFF = {0,1024,1536,1792,1920,1984,2016}.
// Returns global node index and local index of left child (-1 for leaves).
__device__ __forceinline__ void gmap_local(int li, int blk, int& gi, int& childLocal) {
  int l, LOFFv, OFFSv, CLO;
  if      (li < 1024) { l = 0; LOFFv = 0;    OFFSv = 0;      CLO = -1;   }
  else if (li < 1536) { l = 1; LOFFv = 1024; OFFSv = 65536;  CLO = 0;    }
  else if (li < 1792) { l = 2; LOFFv = 1536; OFFSv = 98304;  CLO = 1024; }
  else if (li < 1920) { l = 3; LOFFv = 1792; OFFSv = 114688; CLO = 1536; }
  else if (li < 1984) { l = 4; LOFFv = 1920; OFFSv = 122880; CLO = 1792; }
  else if (li < 2016) { l = 5; LOFFv = 1984; OFFSv = 126976; CLO = 1920; }
  else                { l = 6; LOFFv = 2016; OFFSv = 129024; CLO = 1984; }
  const int j = li - LOFFv;
  gi = OFFSv + (blk << (10 - l)) + j;
  childLocal = (CLO < 0) ? -1 : (CLO + 2 * j);
}

// Top region (levels 7..16, global [130048, 131071), handled by block 0).
// tj in [0,1023) with per-level offsets TOFF = {0,512,768,896,960,992,1008,1016,1020,1022}.
// coff = child level's offset within the top arrays; -1 => children are level-6 (g6 buffer).
__device__ __forceinline__ void tmap(int tj, int& toff, int& coff) {
  if      (tj < 512)  { toff = 0;    coff = -1;   }
  else if (tj < 768)  { toff = 512;  coff = 0;    }
  else if (tj < 896)  { toff = 768;  coff = 512;  }
  else if (tj < 960)  { toff = 896;  coff = 768;  }
  else if (tj < 992)  { toff = 960;  coff = 896;  }
  else if (tj < 1008) { toff = 992;  coff = 960;  }
  else if (tj < 1016) { toff = 1008; coff = 992;  }
  else if (tj < 1020) { toff = 1016; coff = 1008; }
  else if (tj < 1022) { toff = 1020; coff = 1016; }
  else                { toff = 1022; coff = 1020; }
}

// --- per-node Muskingum-Cunge coefficients ----------------------------------
__device__ __forceinline__ void mc_coeffs(float q, float qp_t,
                                          float n_raw, float qs_raw, float len,
                                          float slope, float tw, float ss, float x,
                                          float it, float& bOut, float& c1Out) {
  const float n   = 0.01f + n_raw * 0.34f;          // N_LO + raw*(N_HI-N_LO)
  const float qs  = 3.0f * qs_raw;                  // QS_LO + raw*(QS_HI-QS_LO)
  const float s0  = fmaxf(slope, 1.0e-4f);
  const float rs  = sqrtf(s0);
  const float num = q * n * (qs + 1.0f);
  const float den = 21.0f * rs + 1.0e-8f;           // P_SPATIAL*sqrt(s0)+eps
  const float depth  = fmaxf(powf(num / den, 3.0f / (5.0f + 3.0f * qs)), 0.01f);
  const float bottom = fmaxf(tw - 2.0f * ss * depth, 0.1f);
  const float area   = (tw + bottom) * depth * 0.5f;
  const float wetted = bottom + 2.0f * depth * sqrtf(1.0f + ss * ss);
  const float R      = area / wetted;
  float v = (1.0f / n) * powf(R, 2.0f / 3.0f) * rs;
  v = fminf(fmaxf(v, 0.3f), 15.0f) * (5.0f / 3.0f);
  const float k     = len / v;
  const float denom = 2.0f * k * (1.0f - x) + 3600.0f;
  const float c2 = (3600.0f + 2.0f * k * x) / denom;
  const float c3 = (2.0f * k * (1.0f - x) - 3600.0f) / denom;
  const float c4 = 7200.0f / denom;
  c1Out = (3600.0f - 2.0f * k * x) / denom;         // sweep: q1[p]=b[p]+c1[p]*childsum
  const float qpc = fmaxf(qp_t, Q_LB_F);
  bOut = c2 * it + c3 * q + c4 * qpc;
}

// --- epoch-based grid-wide barrier (all NBLK blocks co-resident) ------------
__device__ __forceinline__ void grid_barrier(unsigned* cnt, unsigned* flag, unsigned epoch) {
  __threadfence();  // release prior global writes (g6 publish)
  if (threadIdx.x == 0) {
    const unsigned prev =
        __hip_atomic_fetch_add(cnt, 1u, __ATOMIC_ACQ_REL, __HIP_MEMORY_SCOPE_AGENT);
    if (prev == epoch * (unsigned)NBLK - 1u) {
      __hip_atomic_store(flag, epoch, __ATOMIC_RELEASE, __HIP_MEMORY_SCOPE_AGENT);
    } else {
      while (__hip_atomic_load(flag, __ATOMIC_ACQUIRE, __HIP_MEMORY_SCOPE_AGENT) < epoch) {
        __builtin_amdgcn_s_sleep(2);
      }
    }
  }
  __syncthreads();
  __threadfence();  // acquire for all threads in the block
}

__global__ void init_sync_kernel(unsigned* syncbuf) {
  syncbuf[0] = 0u;  // arrive counter
  syncbuf[1] = 0u;  // release flag
}

// ============================================================================
__global__ void __launch_bounds__(NTHR, 1)
river_route(const float* __restrict__ qp,      // (336, 131071)
            const float* __restrict__ n_raw,
            const float* __restrict__ qs_raw,
            const float* __restrict__ len,
            const float* __restrict__ slope,
            const float* __restrict__ tw,
            const float* __restrict__ ss,
            const float* __restrict__ xs,
            float* __restrict__ out,           // 336 floats
            unsigned* __restrict__ syncbuf,    // [cnt, flag]
            float* __restrict__ g6)            // double-buffered level-6 q1: 2*1024
{
  // LDS-resident state (persists across all steps): ~60KB total
  __shared__ float q1s[2032];        // levels 0..6 slice state
  __shared__ float bA [2032];
  __shared__ float c1A[2032];
  __shared__ float q1t[1023];        // top levels 7..16 (block 0 only)
  __shared__ float bT [1023];
  __shared__ float c1T[1023];
  __shared__ float stage[2][3072];   // q_prime row staging (async prefetch)

  const int  blk = blockIdx.x;
  const int  tid = threadIdx.x;
  const bool top = (blk == 0);
  unsigned epoch = 0;

  // ---- init: carry state = q_prime row 0 -----------------------------------
  for (int li = tid; li < 2032; li += NTHR) {
    int gi, cl; gmap_local(li, blk, gi, cl);
    q1s[li] = qp[gi];
  }
  if (top) {
    for (int tj = tid; tj < 1023; tj += NTHR) q1t[tj] = qp[130048 + tj];
    if (tid == 0) out[0] = fmaxf(qp[131070], Q_LB_F);
  }
  __syncthreads();
  // publish level-6 state into the "prev" parity buffer for step 0
  for (int j = tid; j < 16; j += NTHR) g6[1024 + blk * 16 + j] = q1s[2016 + j];

  // ---- async prefetch of q_prime row 0 into stage[0] -----------------------
  for (int li = tid; li < 2032; li += NTHR) {
    int gi, cl; gmap_local(li, blk, gi, cl);
    async_f32(qp + gi, &stage[0][li]);
  }
  if (top) {
    for (int tj = tid; tj < 1023; tj += NTHR)
      async_f32(qp + 130048 + tj, &stage[0][2032 + tj]);
  }

  grid_barrier(syncbuf, syncbuf + 1, ++epoch);

  // ---- time loop ----------------------------------------------------------
  for (int step = 0; step < NSTEPS; ++step) {
    const int cur = step & 1, nxt = cur ^ 1;

    wait_async_lds();       // stage[cur] ready (per-wave ASYNCcnt drain)
    __syncthreads();

    // prefetch next row while we compute this step
    if (step + 1 < NSTEPS) {
      const float* row = qp + (size_t)(step + 1) * (size_t)NNODES;
      for (int li = tid; li < 2032; li += NTHR) {
        int gi, cl; gmap_local(li, blk, gi, cl);
        async_f32(row + gi, &stage[nxt][li]);
      }
      if (top) {
        for (int tj = tid; tj < 1023; tj += NTHR)
          async_f32(row + 130048 + tj, &stage[nxt][2032 + tj]);
      }
    }

    // Phase A: elementwise coefficients (levels 0..6) using OLD q1s state
    for (int li = tid; li < 2032; li += NTHR) {
      int gi, cl; gmap_local(li, blk, gi, cl);
      const float q  = fmaxf(q1s[li], Q_LB_F);
      float it = 0.0f;
      if (cl >= 0)
        it = fmaxf(q1s[cl], Q_LB_F) + fmaxf(q1s[cl + 1], Q_LB_F);
      mc_coeffs(q, stage[cur][li],
                n_raw[gi], qs_raw[gi], len[gi], slope[gi], tw[gi], ss[gi], xs[gi],
                it, bA[li], c1A[li]);
    }
    if (top) {
      const float* g6p = g6 + nxt * 1024;   // previous step's level-6 state
      for (int tj = tid; tj < 1023; tj += NTHR) {
        int toff, coff; tmap(tj, toff, coff);
        const int  j  = tj - toff;
        const float q = fmaxf(q1t[tj], Q_LB_F);
        float it;
        if (coff < 0)
          it = fmaxf(g6p[2 * j], Q_LB_F) + fmaxf(g6p[2 * j + 1], Q_LB_F);
        else
          it = fmaxf(q1t[coff + 2 * j], Q_LB_F) + fmaxf(q1t[coff + 2 * j + 1], Q_LB_F);
        const int gi = 130048 + tj;
        mc_coeffs(q, stage[cur][2032 + tj],
                  n_raw[gi], qs_raw[gi], len[gi], slope[gi], tw[gi], ss[gi], xs[gi],
                  it, bT[tj], c1T[tj]);
      }
    }
    __syncthreads();

    // Local sweep levels 0..6 (unclamped intermediates, per reference)
    for (int li = tid; li < 1024; li += NTHR) q1s[li] = bA[li];
    __syncthreads();
    {
      const int loffA[6] = {1024, 1536, 1792, 1920, 1984, 2016};
      const int cloA [6] = {0,    1024, 1536, 1792, 1920, 1984};
      const int cntA [6] = {512,  256,  128,  64,   32,   16};
#pragma unroll
      for (int lv = 0; lv < 6; ++lv) {
        for (int j = tid; j < cntA[lv]; j += NTHR) {
          const int p = loffA[lv] + j, c = cloA[lv] + 2 * j;
          q1s[p] = bA[p] + c1A[p] * (q1s[c] + q1s[c + 1]);
        }
        __syncthreads();
      }
    }
    // publish fresh level-6 values for block 0's top sweep
    for (int j = tid; j < 16; j += NTHR) g6[cur * 1024 + blk * 16 + j] = q1s[2016 + j];

    grid_barrier(syncbuf, syncbuf + 1, ++epoch);   // one barrier per step

    // Phase B: block 0 sweeps levels 7..16 and emits the root
    if (top) {
      const float* g6c = g6 + cur * 1024;
      for (int j = tid; j < 512; j += NTHR)
        q1t[j] = bT[j] + c1T[j] * (g6c[2 * j] + g6c[2 * j + 1]);
      __syncthreads();
      const int toffB[9] = {512, 768, 896, 960, 992, 1008, 1016, 1020, 1022};
      const int cloB [9] = {0,   512, 768, 896, 960, 992,  1008, 1016, 1020};
      const int cntB [9] = {256, 128, 64,  32,  16,  8,    4,    2,    1};
#pragma unroll
      for (int lv = 0; lv < 9; ++lv) {
        for (int j = tid; j < cntB[lv]; j += NTHR) {
          const int p = toffB[lv] + j, c = cloB[lv] + 2 * j;
          q1t[p] = bT[p] + c1T[p] * (q1t[c] + q1t[c + 1]);
        }
        __syncthreads();
      }
      if (tid == 0) out[step + 1] = fmaxf(q1t[1022], Q_LB_F);
    }
  }
}

// ============================================================================
extern "C" void kernel_launch(void* const* d_in, const int* in_sizes, int n_in,
                              void* d_out, int out_size, void* d_ws, size_t ws_size,
                              hipStream_t stream) {
  (void)in_sizes; (void)n_in; (void)out_size; (void)ws_size;
  const float* qp    = (const float*)d_in[0];
  const float* nraw  = (const float*)d_in[1];
  const float* qsraw = (const float*)d_in[2];
  const float* len   = (const float*)d_in[3];
  const float* slp   = (const float*)d_in[4];
  const float* tw    = (const float*)d_in[5];
  const float* ss    = (const float*)d_in[6];
  const float* xs    = (const float*)d_in[7];
  float*    out     = (float*)d_out;
  unsigned* syncbuf = (unsigned*)d_ws;
  float*    g6      = (float*)((char*)d_ws + 256);  // 2*1024 floats

  init_sync_kernel<<<1, 1, 0, stream>>>(syncbuf);
  river_route<<<NBLK, NTHR, 0, stream>>>(qp, nraw, qsraw, len, slp, tw, ss, xs,
                                         out, syncbuf, g6);
}